// LightGCN_12352325943854
// MI455X (gfx1250) — compile-verified
//
#include <hip/hip_runtime.h>
#include <stdint.h>

#define AS1 __attribute__((address_space(1)))
#define AS3 __attribute__((address_space(3)))
#define DIMH 64

// ---------------- CDNA5 async-copy helpers (gfx1250) ----------------
// Per-lane async global->LDS copy, tracked by ASYNCcnt (not LOADcnt).
__device__ __forceinline__ void async_load_b32_to_lds(const int* gptr, uint32_t lds_byte_off) {
#if __has_builtin(__builtin_amdgcn_global_load_async_to_lds_b32)
  __builtin_amdgcn_global_load_async_to_lds_b32(
      (AS1 int*)(uintptr_t)gptr, (AS3 int*)(uintptr_t)lds_byte_off, 0, 0);
#else
  asm volatile("global_load_async_to_lds_b32 %0, %1, off"
               :: "v"(lds_byte_off), "v"((unsigned long long)(uintptr_t)gptr)
               : "memory");
#endif
}

__device__ __forceinline__ void wait_asynccnt_le2() {
#if __has_builtin(__builtin_amdgcn_s_wait_asynccnt)
  __builtin_amdgcn_s_wait_asynccnt(2);
#else
  asm volatile("s_wait_asynccnt 2" ::: "memory");
#endif
}
__device__ __forceinline__ void wait_asynccnt_0() {
#if __has_builtin(__builtin_amdgcn_s_wait_asynccnt)
  __builtin_amdgcn_s_wait_asynccnt(0);
#else
  asm volatile("s_wait_asynccnt 0" ::: "memory");
#endif
}

// Hardware no-CAS f32 atomic add (global_atomic_add_f32).
__device__ __forceinline__ void atomAddF32(float* p, float v) {
  unsafeAtomicAdd(p, v);
}

// ---------------- Kernels ----------------

// Copy x into ping buffer, write both passthrough output copies, seed the
// "final" output accumulator with x, and zero the degree array.
__global__ void __launch_bounds__(256) lgcn_init(
    const float4* __restrict__ u4, const float4* __restrict__ it4,
    float4* __restrict__ bufA, float4* __restrict__ out,
    float* __restrict__ deg, int NU16, int NI16, int N)
{
  const int total = NU16 + NI16;
  const int stride = gridDim.x * blockDim.x;
  for (int i = blockIdx.x * blockDim.x + threadIdx.x; i < total; i += stride) {
    const float4 v = (i < NU16) ? u4[i] : it4[i - NU16];
    bufA[i] = v;
    size_t f, c;
    if (i < NU16) { f = (size_t)i;                       c = (size_t)NU16 + i; }
    else          { f = (size_t)2 * NU16 + (i - NU16);   c = (size_t)2 * NU16 + NI16 + (i - NU16); }
    out[f] = v;   // accumulator seeded with x (layer-0 term of `total`)
    out[c] = v;   // passthrough copy of the raw embedding
    if (i < N) deg[i] = 0.0f;
  }
}

// Degree of each dst node. Non-temporal edge loads keep the 16MB stream out
// of L2 so the feature buffers stay resident.
__global__ void __launch_bounds__(256) lgcn_deg(
    const int* __restrict__ dst, float* __restrict__ deg, int E)
{
  const int stride = gridDim.x * blockDim.x;
  for (int e = blockIdx.x * blockDim.x + threadIdx.x; e < E; e += stride) {
    const int t = __builtin_nontemporal_load(dst + e);
    atomAddF32(deg + t, 1.0f);
  }
}

__global__ void __launch_bounds__(256) lgcn_dinv(float* __restrict__ deg, int N) {
  const int stride = gridDim.x * blockDim.x;
  for (int i = blockIdx.x * blockDim.x + threadIdx.x; i < N; i += stride) {
    const float d = deg[i];
    deg[i] = (d > 0.0f) ? rsqrtf(d) : 0.0f;   // deg is integer-valued, >=1 when >0
  }
}

__global__ void __launch_bounds__(256) lgcn_zero4(float4* __restrict__ p, int n4) {
  const int stride = gridDim.x * blockDim.x;
  const float4 z = make_float4(0.f, 0.f, 0.f, 0.f);
  for (int i = blockIdx.x * blockDim.x + threadIdx.x; i < n4; i += stride) p[i] = z;
}

// One wave32 per edge: lanes cover the 64 feature dims (2 floats each).
// Edge (src,dst) pairs are double-buffered into LDS with async copies.
__global__ void __launch_bounds__(256) lgcn_prop(
    const int* __restrict__ src, const int* __restrict__ dst,
    const float* __restrict__ dinv, const float* __restrict__ acc,
    float* __restrict__ nxt, int E)
{
  __shared__ int lsrc[8][2][32];
  __shared__ int ldst[8][2][32];

  const int lane = (int)(threadIdx.x & 31u);
  const int wv   = (int)(threadIdx.x >> 5);
  const int wpb  = (int)(blockDim.x >> 5);
  const int gw   = blockIdx.x * wpb + wv;
  const int nw   = gridDim.x * wpb;
  int base = gw * 32;
  if (base >= E) return;                 // wave-uniform
  const int stride = nw * 32;

  const uint32_t off_s[2] = { (uint32_t)(uintptr_t)&lsrc[wv][0][lane],
                              (uint32_t)(uintptr_t)&lsrc[wv][1][lane] };
  const uint32_t off_d[2] = { (uint32_t)(uintptr_t)&ldst[wv][0][lane],
                              (uint32_t)(uintptr_t)&ldst[wv][1][lane] };

  // Prologue: stage 0 in flight.
  if (base + lane < E) {
    async_load_b32_to_lds(src + base + lane, off_s[0]);
    async_load_b32_to_lds(dst + base + lane, off_d[0]);
  }

  int buf = 0;
  for (; base < E; base += stride) {
    const int  nb        = base + stride;
    const bool have_next = nb < E;
    if (have_next && (nb + lane < E)) {
      async_load_b32_to_lds(src + nb + lane, off_s[buf ^ 1]);
      async_load_b32_to_lds(dst + nb + lane, off_d[buf ^ 1]);
    }
    // 2 ops per stage: with next stage in flight, wait ASYNCcnt<=2 so the
    // current stage's pair (oldest, completes in order) has landed in LDS.
    if (have_next) wait_asynccnt_le2(); else wait_asynccnt_0();

    const int  cnt = (E - base < 32) ? (E - base) : 32;
    const int* cs  = lsrc[wv][buf];
    const int* cd  = ldst[wv][buf];

    int s = cs[0], t = cd[0];
    for (int e = 0; e < cnt; ++e) {
      int s2 = 0, t2 = 0;
      if (e + 1 < cnt) {
        s2 = cs[e + 1]; t2 = cd[e + 1];
        __builtin_prefetch(acc + (size_t)s2 * DIMH, 0, 0);  // global_prefetch_b8
        __builtin_prefetch(nxt + (size_t)t2 * DIMH, 1, 0);
      }
      const float  nrm = dinv[s] * dinv[t];
      const float2 v   = *reinterpret_cast<const float2*>(acc + (size_t)s * DIMH + lane * 2);
      float* op = nxt + (size_t)t * DIMH + (size_t)(lane * 2);
      atomAddF32(op,     v.x * nrm);
      atomAddF32(op + 1, v.y * nrm);
      s = s2; t = t2;
    }
    buf ^= 1;
  }
}

// total += acc_layer, written into the split [users_final | ... | items_final] layout.
__global__ void __launch_bounds__(256) lgcn_accum(
    const float4* __restrict__ nxt, float4* __restrict__ out, int NU16, int NI16)
{
  const int total = NU16 + NI16;
  const int stride = gridDim.x * blockDim.x;
  for (int i = blockIdx.x * blockDim.x + threadIdx.x; i < total; i += stride) {
    const size_t f = (i < NU16) ? (size_t)i : (size_t)2 * NU16 + (i - NU16);
    float4 a = out[f];
    const float4 b = nxt[i];
    a.x += b.x; a.y += b.y; a.z += b.z; a.w += b.w;
    out[f] = a;
  }
}

__global__ void __launch_bounds__(256) lgcn_scale(
    float4* __restrict__ out, int NU16, int NI16, float s)
{
  const int total = NU16 + NI16;
  const int stride = gridDim.x * blockDim.x;
  for (int i = blockIdx.x * blockDim.x + threadIdx.x; i < total; i += stride) {
    const size_t f = (i < NU16) ? (size_t)i : (size_t)2 * NU16 + (i - NU16);
    float4 a = out[f];
    a.x *= s; a.y *= s; a.z *= s; a.w *= s;
    out[f] = a;
  }
}

// ---------------- Host launcher ----------------
extern "C" void kernel_launch(void* const* d_in, const int* in_sizes, int n_in,
                              void* d_out, int out_size, void* d_ws, size_t ws_size,
                              hipStream_t stream) {
  (void)n_in; (void)out_size; (void)ws_size;
  const float* users = (const float*)d_in[0];
  const float* items = (const float*)d_in[1];
  const int*   edges = (const int*)d_in[2];

  const int NU = in_sizes[0] / DIMH;
  const int NI = in_sizes[1] / DIMH;
  const int E  = in_sizes[2] / 2;
  const int N  = NU + NI;
  const int NU16 = NU * (DIMH / 4);
  const int NI16 = NI * (DIMH / 4);

  const int* srcp = edges;
  const int* dstp = edges + E;

  // Workspace: [deg/dinv : N floats (padded)] [bufA : N*64] [bufB : N*64]
  float* deg  = (float*)d_ws;
  const size_t degPad = ((size_t)N + 255) & ~(size_t)255;
  float* bufA = deg + degPad;
  float* bufB = bufA + (size_t)N * DIMH;
  float* out  = (float*)d_out;

  const int B = 256;
  lgcn_init<<<4096, B, 0, stream>>>((const float4*)users, (const float4*)items,
                                    (float4*)bufA, (float4*)out, deg, NU16, NI16, N);
  lgcn_deg<<<2048, B, 0, stream>>>(dstp, deg, E);
  lgcn_dinv<<<512, B, 0, stream>>>(deg, N);

  float* acc = bufA;
  float* nxt = bufB;
  for (int l = 0; l < 4; ++l) {
    lgcn_zero4<<<2048, B, 0, stream>>>((float4*)nxt, NU16 + NI16);
    lgcn_prop<<<2048, B, 0, stream>>>(srcp, dstp, deg, acc, nxt, E);
    lgcn_accum<<<2048, B, 0, stream>>>((const float4*)nxt, (float4*)out, NU16, NI16);
    float* tmp = acc; acc = nxt; nxt = tmp;
  }
  lgcn_scale<<<2048, B, 0, stream>>>((float4*)out, NU16, NI16, 1.0f / 25.0f);
}